// SAGE_40999757808109
// MI455X (gfx1250) — compile-verified
//
#include <hip/hip_runtime.h>
#include <math.h>

// ---------------------------------------------------------------------------
// Problem constants (from reference):
//   N0=3072000 N1=204800 N2=20480 N3=4096 ; E0=3072000 E1=204800 E2=20480
//   D_IN=100 D_H=256 D_OUT=47
// ---------------------------------------------------------------------------
#define N0c  3072000
#define N1c  204800
#define N2c  20480
#define N3c  4096
#define E0c  3072000
#define E1c  204800
#define E2c  20480
#define DINc 100
#define DHc  256
#define DOUTc 47

typedef __attribute__((ext_vector_type(2))) float v2f;
typedef __attribute__((ext_vector_type(8))) float v8f;

// ---------------------------------------------------------------------------
// Edge scatter: one wave (32 lanes) per edge.
//   s[dst]   += x[src]   (D floats, global_atomic_add_f32)
//   cnt[dst] += 1
// ---------------------------------------------------------------------------
__global__ __launch_bounds__(256)
void sage_scatter(const float* __restrict__ x,
                  const int*   __restrict__ src,
                  const int*   __restrict__ dst,
                  float*       __restrict__ s,
                  float*       __restrict__ cnt,
                  int E, int D) {
  int gid  = blockIdx.x * blockDim.x + threadIdx.x;
  int wid  = gid >> 5;
  int lane = threadIdx.x & 31;
  if (wid >= E) return;
  int sr = src[wid];
  int dr = dst[wid];
  const float* xr = x + (long long)sr * D;
  float*       so = s + (long long)dr * D;
  for (int f = lane; f < D; f += 32) {
    atomicAdd(&so[f], xr[f]);
  }
  if (lane == 0) atomicAdd(&cnt[dr], 1.0f);
}

// ---------------------------------------------------------------------------
// Fused SAGE layer GEMM via V_WMMA_F32_16X16X4_F32 (wave32):
//   out[M x N] = act( (s/max(cnt,1)) @ Wl + bl + xin @ Wr )
// Block = 128 threads (4 waves), one 16-row M tile per block.
// Wave w handles N-tiles {w + 4*t | t < NT} (16 cols each), NT compile-time.
// Fragment layouts per CDNA5 ISA tables:
//   A (16x4, from LDS):  lane<16: M=lane,K={k0,k0+1}; lane>=16: M=lane-16,K={k0+2,k0+3}
//   B (4x16, K-major):   VGPR0 = B[k0+(lane>>4)*2][n], VGPR1 = B[k0+(lane>>4)*2+1][n]
//   C/D (16x16):         VGPR v -> row v (lanes 0-15) / v+8 (lanes 16-31), col = lane&15
// GUARD (N not multiple of 16) uses clamp+mask (no EXEC divergence near WMMA).
// ---------------------------------------------------------------------------
template <int K, int N, int NT, bool RELU>
__global__ __launch_bounds__(128)
void sage_gemm_wmma(const float* __restrict__ s,    // M x K segment sums
                    const float* __restrict__ cnt,  // M counts
                    const float* __restrict__ xin,  // M x K self features
                    const float* __restrict__ Wl,   // K x N
                    const float* __restrict__ bl,   // N
                    const float* __restrict__ Wr,   // K x N
                    float* __restrict__ out) {      // M x N
  constexpr bool GUARD = (N & 15) != 0;
  constexpr int  LD    = K + 4;
  __shared__ float As[16 * LD];   // agg tile
  __shared__ float Xs[16 * LD];   // self tile
  const int rowbase = blockIdx.x * 16;

  // Cooperative stage of A tiles (divide-by-count fused here).
  for (int i = threadIdx.x; i < 16 * K; i += 128) {
    int r = i / K;
    int k = i - r * K;
    long long g = (long long)(rowbase + r) * K + k;
    float inv = 1.0f / fmaxf(cnt[rowbase + r], 1.0f);
    As[r * LD + k] = s[g] * inv;
    Xs[r * LD + k] = xin[g];
  }
  __syncthreads();

  const int lane  = threadIdx.x & 31;
  const int wave  = threadIdx.x >> 5;
  const int m     = lane & 15;
  const int koff  = (lane >> 4) << 1;   // 0 or 2
  const int ncol  = lane & 15;
  const int rhalf = (lane >> 4) << 3;   // 0 or 8

  v8f   acc[NT];
  int   col[NT];    // true output column
  int   bcol[NT];   // clamped (always in-bounds) column for loads
  float bmask[NT];  // 1.0 if col valid else 0.0
  const v8f vzero = {0.f, 0.f, 0.f, 0.f, 0.f, 0.f, 0.f, 0.f};
#pragma unroll
  for (int t = 0; t < NT; ++t) {
    acc[t]   = vzero;
    col[t]   = ((wave + 4 * t) << 4) + ncol;
    bcol[t]  = GUARD ? (col[t] < N ? col[t] : N - 1) : col[t];
    bmask[t] = (!GUARD || col[t] < N) ? 1.0f : 0.0f;
  }

  // Phase 0: agg @ Wl ; Phase 1: x_tgt @ Wr — accumulated into same C.
#pragma unroll 1
  for (int phase = 0; phase < 2; ++phase) {
    const float* Ash = phase ? Xs : As;
    const float* B   = phase ? Wr : Wl;
#pragma unroll 2
    for (int k0 = 0; k0 < K; k0 += 4) {
      v2f a;
      a.x = Ash[m * LD + k0 + koff];
      a.y = Ash[m * LD + k0 + koff + 1];
#pragma unroll
      for (int t = 0; t < NT; ++t) {
        v2f b;
        b.x = B[(k0 + koff) * N + bcol[t]];
        b.y = B[(k0 + koff + 1) * N + bcol[t]];
        if (GUARD) { b.x *= bmask[t]; b.y *= bmask[t]; }
        // v_wmma_f32_16x16x4_f32 : D = A(16x4) x B(4x16) + C
        acc[t] = __builtin_amdgcn_wmma_f32_16x16x4_f32(
            /*neg_a=*/false, a, /*neg_b=*/false, b,
            /*c_mod=*/(short)0, acc[t],
            /*reuse_a=*/false, /*reuse_b=*/false);
      }
    }
  }

  // Epilogue: bias (+ ReLU) and store.
#pragma unroll
  for (int t = 0; t < NT; ++t) {
    float bias = bl[bcol[t]];
    bool valid = !GUARD || (col[t] < N);
#pragma unroll
    for (int v = 0; v < 8; ++v) {
      int row   = rowbase + rhalf + v;
      float val = acc[t][v] + bias;
      if (RELU) val = fmaxf(val, 0.0f);
      if (valid) out[(long long)row * N + col[t]] = val;
    }
  }
}

// ---------------------------------------------------------------------------
// log_softmax over rows of C<=64 columns; one wave per row.
// ---------------------------------------------------------------------------
__global__ __launch_bounds__(256)
void log_softmax_rows(const float* __restrict__ logits,
                      float* __restrict__ out, int rows, int C) {
  int gid  = blockIdx.x * blockDim.x + threadIdx.x;
  int wid  = gid >> 5;
  int lane = threadIdx.x & 31;
  if (wid >= rows) return;
  const float* rp = logits + (long long)wid * C;
  float v0 = (lane < C)      ? rp[lane]      : -INFINITY;
  float v1 = (lane + 32 < C) ? rp[lane + 32] : -INFINITY;
  float mx = fmaxf(v0, v1);
  for (int o = 16; o > 0; o >>= 1) mx = fmaxf(mx, __shfl_xor(mx, o, 32));
  float e0 = (lane < C)      ? expf(v0 - mx) : 0.0f;
  float e1 = (lane + 32 < C) ? expf(v1 - mx) : 0.0f;
  float sum = e0 + e1;
  for (int o = 16; o > 0; o >>= 1) sum += __shfl_xor(sum, o, 32);
  float lse = logf(sum) + mx;
  float* op = out + (long long)wid * C;
  if (lane < C)      op[lane]      = v0 - lse;
  if (lane + 32 < C) op[lane + 32] = v1 - lse;
}

// ---------------------------------------------------------------------------
extern "C" void kernel_launch(void* const* d_in, const int* in_sizes, int n_in,
                              void* d_out, int out_size, void* d_ws, size_t ws_size,
                              hipStream_t stream) {
  const float* x    = (const float*)d_in[0];
  const int*   src0 = (const int*)d_in[1];
  const int*   dst0 = (const int*)d_in[2];
  const int*   src1 = (const int*)d_in[3];
  const int*   dst1 = (const int*)d_in[4];
  const int*   src2 = (const int*)d_in[5];
  const int*   dst2 = (const int*)d_in[6];
  const float* Wl0  = (const float*)d_in[7];
  const float* bl0  = (const float*)d_in[8];
  const float* Wr0  = (const float*)d_in[9];
  const float* Wl1  = (const float*)d_in[10];
  const float* bl1  = (const float*)d_in[11];
  const float* Wr1  = (const float*)d_in[12];
  const float* Wl2  = (const float*)d_in[13];
  const float* bl2  = (const float*)d_in[14];
  const float* Wr2  = (const float*)d_in[15];

  // Workspace layout (floats). Atomically-accumulated region first (zeroed).
  float* ws = (float*)d_ws;
  size_t off = 0;
  float* s0 = ws + off; off += (size_t)N1c * DINc;   // 20,480,000
  float* c0 = ws + off; off += (size_t)N1c;
  float* s1 = ws + off; off += (size_t)N2c * DHc;    //  5,242,880
  float* c1 = ws + off; off += (size_t)N2c;
  float* s2 = ws + off; off += (size_t)N3c * DHc;    //  1,048,576
  float* c2 = ws + off; off += (size_t)N3c;
  size_t zeroFloats = off;                           // ~108 MB memset
  float* h1 = ws + off; off += (size_t)N1c * DHc;    // 210 MB
  float* h2 = ws + off; off += (size_t)N2c * DHc;    //  21 MB
  float* lg = ws + off; off += (size_t)N3c * DOUTc;

  hipMemsetAsync(d_ws, 0, zeroFloats * sizeof(float), stream);

  // ---- Layer 0 : K=100, N=256, relu ----
  {
    int blocks = (int)(((long long)E0c * 32 + 255) / 256);
    sage_scatter<<<blocks, 256, 0, stream>>>(x, src0, dst0, s0, c0, E0c, DINc);
    sage_gemm_wmma<DINc, DHc, 4, true>
        <<<N1c / 16, 128, 0, stream>>>(s0, c0, x, Wl0, bl0, Wr0, h1);
  }
  // ---- Layer 1 : K=256, N=256, relu ----
  {
    int blocks = (int)(((long long)E1c * 32 + 255) / 256);
    sage_scatter<<<blocks, 256, 0, stream>>>(h1, src1, dst1, s1, c1, E1c, DHc);
    sage_gemm_wmma<DHc, DHc, 4, true>
        <<<N2c / 16, 128, 0, stream>>>(s1, c1, h1, Wl1, bl1, Wr1, h2);
  }
  // ---- Layer 2 : K=256, N=47, no relu (tiles 0..3 -> one per wave) ----
  {
    int blocks = (int)(((long long)E2c * 32 + 255) / 256);
    sage_scatter<<<blocks, 256, 0, stream>>>(h2, src2, dst2, s2, c2, E2c, DHc);
    sage_gemm_wmma<DHc, DOUTc, 1, false>
        <<<N3c / 16, 128, 0, stream>>>(s2, c2, h2, Wl2, bl2, Wr2, lg);
  }
  // ---- log_softmax ----
  {
    int blocks = (int)(((long long)N3c * 32 + 255) / 256);
    log_softmax_rows<<<blocks, 256, 0, stream>>>(lg, (float*)d_out, N3c, DOUTc);
  }
}